// LocalGlobalPattern_55490977465133
// MI455X (gfx1250) — compile-verified
//
#include <hip/hip_runtime.h>

typedef __attribute__((ext_vector_type(2))) float v2f;
typedef __attribute__((ext_vector_type(8))) float v8f;

#define SEQ    4096
#define DM     1024
#define IDIM   32
#define NROWS  8192           // B * S
#define WIN    64
#define GKTOP  64

#define RT       16           // query rows per block = one WMMA M tile
#define NWAVE    16           // waves per block (kernel B)
#define THREADS  512
#define CPW      (SEQ / NWAVE)     // 256 columns per wave
#define TILES    (CPW / 16)        // 16 WMMA N tiles per wave
#define CHUNK    (SEQ / THREADS)   // 8 contiguous elems per thread in select/emit

#define AWAVES   8            // waves per block in kernel A (k-split)
#define KSLICE   (DM / AWAVES)     // 128 K values per wave

// ---------------------------------------------------------------------------
// Kernel A: fp32 WMMA projections.  gq = x @ Wq, gk = x @ Wk.
// 256-thread block per 16-row tile; K dimension split across 8 waves so the
// load->WMMA latency chain is hidden by wave parallelism.  Partial tiles are
// reduced 8-way through LDS.  Tiles: t0/t1 = gq cols 0-15/16-31, t2/t3 = gk.
// ---------------------------------------------------------------------------
__global__ void __launch_bounds__(256)
proj_wmma_kernel(const float* __restrict__ x,
                 const float* __restrict__ Wq,
                 const float* __restrict__ Wk,
                 float* __restrict__ gq,
                 float* __restrict__ gk) {
  __shared__ float s_red[AWAVES][4][256];   // 32 KB partial-tile buffer

  const int tid  = threadIdx.x;
  const int w    = tid >> 5;
  const int lane = tid & 31;
  const int hf   = lane >> 4;
  const int l    = lane & 15;
  const int row0 = blockIdx.x * 16;
  const int kbase = w * KSLICE;

  const float2* xr =
      reinterpret_cast<const float2*>(x + (size_t)(row0 + l) * DM);

  v8f c0 = {}; v8f c1 = {}; v8f c2 = {}; v8f c3 = {};

  for (int kk = 0; kk < KSLICE / 4; ++kk) {
    const int k0  = kbase + kk * 4;
    const int ka0 = k0 + 2 * hf;            // lane's K pair: (ka0, ka0+1)

    const float2 av = xr[(k0 >> 1) + hf];   // contiguous float2
    v2f a; a.x = av.x; a.y = av.y;

    v2f b0, b1, b2, b3;
    b0.x = Wq[(size_t)ka0 * IDIM + l];
    b0.y = Wq[(size_t)(ka0 + 1) * IDIM + l];
    b1.x = Wq[(size_t)ka0 * IDIM + 16 + l];
    b1.y = Wq[(size_t)(ka0 + 1) * IDIM + 16 + l];
    b2.x = Wk[(size_t)ka0 * IDIM + l];
    b2.y = Wk[(size_t)(ka0 + 1) * IDIM + l];
    b3.x = Wk[(size_t)ka0 * IDIM + 16 + l];
    b3.y = Wk[(size_t)(ka0 + 1) * IDIM + 16 + l];

    c0 = __builtin_amdgcn_wmma_f32_16x16x4_f32(false, a, false, b0, (short)0, c0, false, false);
    c1 = __builtin_amdgcn_wmma_f32_16x16x4_f32(false, a, false, b1, (short)0, c1, false, false);
    c2 = __builtin_amdgcn_wmma_f32_16x16x4_f32(false, a, false, b2, (short)0, c2, false, false);
    c3 = __builtin_amdgcn_wmma_f32_16x16x4_f32(false, a, false, b3, (short)0, c3, false, false);
  }

  // dump partial tiles (C/D layout: elem = r*32 + lane)
#pragma unroll
  for (int r = 0; r < 8; ++r) {
    s_red[w][0][r * 32 + lane] = c0[r];
    s_red[w][1][r * 32 + lane] = c1[r];
    s_red[w][2][r * 32 + lane] = c2[r];
    s_red[w][3][r * 32 + lane] = c3[r];
  }
  __syncthreads();

  // 8-way reduction; each thread sums one element of each of the 4 tiles
#pragma unroll
  for (int t = 0; t < 4; ++t) {
    float s = 0.f;
#pragma unroll
    for (int ww = 0; ww < AWAVES; ++ww) s += s_red[ww][t][tid];
    const int r   = tid >> 5;
    const int ln  = tid & 31;
    const int ehf = ln >> 4;
    const int el  = ln & 15;
    const int m   = row0 + r + 8 * ehf;
    const int col = el + ((t & 1) ? 16 : 0);
    float* dst = (t < 2) ? gq : gk;
    dst[(size_t)m * IDIM + col] = s;
  }
}

// ---------------------------------------------------------------------------
// Kernel B: 16 query rows per block.  Phase 1 computes the 16x4096 score
// strip with fp32 WMMA (each of 16 waves owns 256 columns as 16 accumulator
// tiles held in VGPRs).  Then, per row: the owning lane-half dumps its
// accumulator row to a 16KB LDS buffer, 512 threads run an exact byte-radix
// select of the 64th-largest fp32 bit pattern (scores >= 0 after ReLU, so
// uint bits are order-preserving), equal values are taken lowest-index-first
// (jax top_k tie-break) via wave-shuffle prefix scans, and the mask row is
// streamed out with float4 stores.
// ---------------------------------------------------------------------------
__global__ void __launch_bounds__(THREADS)
score_topk_mask_kernel(const float* __restrict__ gq,
                       const float* __restrict__ gk,
                       float* __restrict__ out) {
  __shared__ float    s_row[SEQ];      // 16 KB: one score row at a time
  __shared__ unsigned s_hist[256];
  __shared__ unsigned s_bc[2];         // {prefix, remaining}
  __shared__ unsigned s_wtot[NWAVE];

  const int tid  = threadIdx.x;
  const int wid  = tid >> 5;
  const int lane = tid & 31;
  const int hf   = lane >> 4;
  const int l16  = lane & 15;

  const int m0 = blockIdx.x * RT;          // global row-tile base (b*4096+i)
  const int b  = m0 >> 12;
  const float* gkb = gk + (size_t)b * SEQ * IDIM;

  // ---------------- Phase 1: WMMA scores, ReLU in registers ----------------
  v2f a[8];
  {
    const float2* gqr2 =
        reinterpret_cast<const float2*>(gq + (size_t)(m0 + l16) * IDIM);
#pragma unroll
    for (int kk = 0; kk < 8; ++kk) {
      const float2 av = gqr2[2 * kk + hf];
      a[kk].x = av.x;
      a[kk].y = av.y;
    }
  }

  v8f acc[TILES];
#pragma unroll
  for (int t = 0; t < TILES; ++t) acc[t] = (v8f){};

  const int colbase = wid * CPW;
#pragma unroll
  for (int t = 0; t < TILES; ++t) {
    const float2* gkr2 = reinterpret_cast<const float2*>(
        gkb + (size_t)(colbase + t * 16 + l16) * IDIM);
#pragma unroll
    for (int kk = 0; kk < 8; ++kk) {
      const float2 bv = gkr2[2 * kk + hf];
      v2f bf;
      bf.x = bv.x;
      bf.y = bv.y;
      acc[t] = __builtin_amdgcn_wmma_f32_16x16x4_f32(
          false, a[kk], false, bf, (short)0, acc[t], false, false);
    }
  }
#pragma unroll
  for (int t = 0; t < TILES; ++t)
#pragma unroll
    for (int r = 0; r < 8; ++r)
      acc[t][r] = acc[t][r] > 0.f ? acc[t][r] : 0.f;

  const float NEGINF = __uint_as_float(0xFF800000u);

  // ---------------- Phase 2: per-row select + emit ----------------
  for (int m = 0; m < RT; ++m) {
    __syncthreads();   // previous row's readers are done with s_row

    // Dump row m: lanes of half m>>3 hold it in accumulator element m&7.
    if (hf == (m >> 3)) {
      switch (m & 7) {
#define DUMP_CASE(R)                                                     \
  case R: {                                                              \
    _Pragma("unroll") for (int t = 0; t < TILES; ++t)                    \
        s_row[colbase + t * 16 + l16] = acc[t][R];                       \
  } break;
        DUMP_CASE(0) DUMP_CASE(1) DUMP_CASE(2) DUMP_CASE(3)
        DUMP_CASE(4) DUMP_CASE(5) DUMP_CASE(6) DUMP_CASE(7)
#undef DUMP_CASE
      }
    }
    __syncthreads();

    // ---- exact top-64 threshold: 4-round byte radix select on fp32 bits ----
    unsigned prefix = 0u, remaining = GKTOP;
    for (int shift = 24; shift >= 0; shift -= 8) {
      if (tid < 256) s_hist[tid] = 0u;
      __syncthreads();
      const unsigned pmask = (shift == 24) ? 0u : (0xFFFFFFFFu << (shift + 8));
#pragma unroll
      for (int k = 0; k < CHUNK; ++k) {
        const unsigned u = __float_as_uint(s_row[tid * CHUNK + k]);
        if ((u & pmask) == prefix)
          atomicAdd(&s_hist[(u >> shift) & 0xFFu], 1u);
      }
      __syncthreads();
      if (wid == 0) {
        // wave 0: lane owns bins [8*lane, 8*lane+8); suffix-sum across lanes
        unsigned tot = 0u;
#pragma unroll
        for (int v = 0; v < 8; ++v) tot += s_hist[lane * 8 + v];
        unsigned sfx = tot;                       // inclusive suffix (desc)
#pragma unroll
        for (int d = 1; d < 32; d <<= 1) {
          const unsigned o = __shfl_down(sfx, (unsigned)d, 32);
          if (lane + d < 32) sfx += o;
        }
        const unsigned above = sfx - tot;         // sum of strictly higher groups
        if (above < remaining && remaining <= sfx) {   // unique lane
          unsigned cum = above;
          int pick = lane * 8;
          for (int v = lane * 8 + 7; v >= lane * 8; --v) {
            const unsigned c = s_hist[v];
            if (cum + c >= remaining) { pick = v; break; }
            cum += c;
          }
          s_bc[0] = prefix | ((unsigned)pick << shift);
          s_bc[1] = remaining - cum;
        }
      }
      __syncthreads();
      prefix    = s_bc[0];
      remaining = s_bc[1];
    }
    const unsigned T       = prefix;     // exact 64th-largest bit pattern
    const unsigned need_eq = remaining;  // #(==T) to keep, lowest index first

    // ---- rank of ==T elements (index order) via shuffle prefix scans ----
    const int j0 = tid * CHUNK;
    unsigned eqc = 0u;
#pragma unroll
    for (int k = 0; k < CHUNK; ++k)
      eqc += (__float_as_uint(s_row[j0 + k]) == T) ? 1u : 0u;
    unsigned incl = eqc;
#pragma unroll
    for (int d = 1; d < 32; d <<= 1) {
      const unsigned o = __shfl_up(incl, (unsigned)d, 32);
      if (lane >= d) incl += o;
    }
    if (lane == 31) s_wtot[wid] = incl;
    __syncthreads();
    unsigned rank = incl - eqc;          // exclusive within wave
    for (int w = 0; w < wid; ++w) rank += s_wtot[w];

    // ---- emit mask row ----
    const int gi = m0 + m;
    const int i  = gi & (SEQ - 1);
    float* orow = out + (size_t)gi * SEQ;
    float vals[CHUNK];
#pragma unroll
    for (int k = 0; k < CHUNK; ++k) {
      const int j = j0 + k;
      const unsigned u = __float_as_uint(s_row[j]);
      bool sel = (u > T);
      if (u == T) { sel = sel || (rank < need_eq); ++rank; }
      const bool band = (j <= i) && (j > i - WIN);
      vals[k] = (band || sel) ? 0.0f : NEGINF;
    }
    float4* o4 = reinterpret_cast<float4*>(orow + j0);
    o4[0] = make_float4(vals[0], vals[1], vals[2], vals[3]);
    o4[1] = make_float4(vals[4], vals[5], vals[6], vals[7]);
  }
}

extern "C" void kernel_launch(void* const* d_in, const int* in_sizes, int n_in,
                              void* d_out, int out_size, void* d_ws, size_t ws_size,
                              hipStream_t stream) {
  (void)in_sizes; (void)n_in; (void)out_size; (void)ws_size;
  const float* x  = (const float*)d_in[0];   // [2, 4096, 1024]
  const float* Wq = (const float*)d_in[1];   // [1024, 32]
  const float* Wk = (const float*)d_in[2];   // [1024, 32]
  float* out = (float*)d_out;                // [2, 1, 4096, 4096]

  float* gq = (float*)d_ws;                  // [8192, 32]
  float* gk = gq + (size_t)NROWS * IDIM;     // [8192, 32]  (2 MB total)

  proj_wmma_kernel<<<NROWS / 16, 256, 0, stream>>>(x, Wq, Wk, gq, gk);
  score_topk_mask_kernel<<<NROWS / RT, THREADS, 0, stream>>>(gq, gk, out);
}